// Plane2Depth_function_2104533975847
// MI455X (gfx1250) — compile-verified
//
#include <hip/hip_runtime.h>

// Plane2Depth for MI455X (gfx1250, wave32).
// feat: (16, 4, 192, 192) f32 ; W: (4,4) f32 ; out: (16, 1, 768, 768) f32.
// Memory-bound (~47 MB traffic => ~2us at 23.3 TB/s). The 4x4 channel mix
// rides the matrix pipe via V_WMMA_F32_16X16X4_F32 (full f32 precision).
// VALU kept lean: v_rsq_f32 + v_rcp_f32 instead of IEEE divide expansions,
// 32-bit address arithmetic so loads/stores use saddr+voffset forms.

typedef __attribute__((ext_vector_type(2))) float v2f;
typedef __attribute__((ext_vector_type(8))) float v8f;

#define B_      16
#define CH_     4
#define H_      192
#define W_      192
#define HW_     (H_ * W_)          // 36864, channel stride in feat
#define UP_     4
#define OW_     (W_ * UP_)         // 768
#define OHW_    (OW_ * OW_)        // 589824, batch stride in out
#define PIX_PER_CHUNK 16
#define CHUNKS_PER_ROW (W_ / PIX_PER_CHUNK)          // 12
#define TOTAL_CHUNKS   (B_ * H_ * CHUNKS_PER_ROW)    // 36864
#define WAVES_PER_BLOCK 8

__global__ __launch_bounds__(256)
void plane2depth_wmma_kernel(const float* __restrict__ feat,
                             const float* __restrict__ Wm,
                             float* __restrict__ out)
{
    const int lane  = threadIdx.x & 31;
    const int chunk = blockIdx.x * WAVES_PER_BLOCK + (threadIdx.x >> 5);

    // ---- A operand: 16x4 f32 = W rows replicated at M=0..3 and M=8..11,
    // zeros elsewhere. ISA layout: lane L holds row M = L%16; lanes 0-15
    // carry K=0 (a.x), K=1 (a.y); lanes 16-31 carry K=2 (a.x), K=3 (a.y).
    const int m   = lane & 15;
    const int kHi = (lane >> 4) << 1;      // 0 for lanes 0-15, 2 for lanes 16-31
    const int m4  = m & 7;                 // fold M=8..11 onto W rows 0..3
    v2f a;
    a.x = (m4 < 4) ? Wm[m4 * 4 + kHi]     : 0.0f;
    a.y = (m4 < 4) ? Wm[m4 * 4 + kHi + 1] : 0.0f;

    const int pix  = lane & 15;            // pixel column within 16-pixel chunk
    const int rSel = (lane >> 4) << 1;     // this lane writes upsample rows rSel, rSel+1

    const int wBlk = chunk % CHUNKS_PER_ROW;
    const int h    = (chunk / CHUNKS_PER_ROW) % H_;
    const int b    =  chunk / (CHUNKS_PER_ROW * H_);
    const int w    = wBlk * PIX_PER_CHUNK + pix;

    // ---- B operand: 4x16 f32 (channels x pixels). Lanes 0-15 supply K=0,1;
    // lanes 16-31 supply K=2,3 for pixel N = lane%16. All offsets fit 32 bits.
    const int inOff = b * (CH_ * HW_) + kHi * HW_ + h * W_ + w;
    v2f bb;
    bb.x = feat[inOff];          // K = kHi
    bb.y = feat[inOff + HW_];    // K = kHi+1  (imm offset 147456 bytes)

    // D[m][n] = plane_eq[m][pixel n]; with duplicated A rows, c[0..3] holds
    // (p,q,r,s) of pixel lane%16 on ALL 32 lanes.
    v8f c = {};
    c = __builtin_amdgcn_wmma_f32_16x16x4_f32(
            /*neg_a=*/false, a, /*neg_b=*/false, bb,
            /*c_mod=*/(short)0, c, /*reuse_a=*/false, /*reuse_b=*/false);

    const float p = c[0], q = c[1], r = c[2], s = c[3];

    // norm = sqrt(p^2+q^2+r^2+eps); fold sn = s*norm into the plane coeffs:
    // disp = (p*u + q*v + r) * rsq * sn  =  ps*u + qs*v + rs
    const float t    = __builtin_fmaf(p, p, __builtin_fmaf(q, q, __builtin_fmaf(r, r, 1e-6f)));
    const float rsq  = __builtin_amdgcn_rsqf(t);   // v_rsq_f32: 1/norm
    const float sn   = s * (t * rsq);              // s * norm
    const float ps   = (p * rsq) * sn;
    const float qs   = (q * rsq) * sn;
    const float rs   = (r * rsq) * sn;

    const int outOff = b * OHW_ + (UP_ * h + rSel) * OW_ + UP_ * w;
    #pragma unroll
    for (int rr = 0; rr < 2; ++rr) {
        const float v    = ((float)(rSel + rr) - 1.5f) * 0.25f;
        const float base = __builtin_fmaf(qs, v, rs);
        float4 o;   // u = (col - 1.5)/4 ; out = 1/max(disp, 0.1), v_rcp_f32
        o.x = __builtin_amdgcn_rcpf(fmaxf(__builtin_fmaf(ps, -0.375f, base), 0.1f));
        o.y = __builtin_amdgcn_rcpf(fmaxf(__builtin_fmaf(ps, -0.125f, base), 0.1f));
        o.z = __builtin_amdgcn_rcpf(fmaxf(__builtin_fmaf(ps,  0.125f, base), 0.1f));
        o.w = __builtin_amdgcn_rcpf(fmaxf(__builtin_fmaf(ps,  0.375f, base), 0.1f));
        *reinterpret_cast<float4*>(out + outOff + rr * OW_) = o;  // 16B-aligned b128
    }
}

extern "C" void kernel_launch(void* const* d_in, const int* in_sizes, int n_in,
                              void* d_out, int out_size, void* d_ws, size_t ws_size,
                              hipStream_t stream) {
    (void)in_sizes; (void)n_in; (void)out_size; (void)d_ws; (void)ws_size;
    const float* feat = (const float*)d_in[0];
    const float* Wm   = (const float*)d_in[1];
    float*       out  = (float*)d_out;

    // Exactly one 16-pixel chunk per wave: 36864 waves = 4608 blocks x 8 waves.
    plane2depth_wmma_kernel<<<TOTAL_CHUNKS / WAVES_PER_BLOCK, 256, 0, stream>>>(feat, Wm, out);
}